// GridEncoder_37744172597961
// MI455X (gfx1250) — compile-verified
//
#include <hip/hip_runtime.h>
#include <math.h>

#define GSZ   96
#define NPTS  (GSZ*GSZ*GSZ)        // 884736
#define NVIEW 8
#define PLANE (GSZ*GSZ)            // 9216

typedef float v2f __attribute__((ext_vector_type(2)));
typedef float v8f __attribute__((ext_vector_type(8)));

// wave-local LDS ordering: all staging hazards are within one wave32,
// so a dscnt wait is sufficient (no block barrier needed).
__device__ __forceinline__ void lds_fence() {
  asm volatile("s_wait_dscnt 0x0" ::: "memory");
}

// ---------------------------------------------------------------------------
// Kernel A: fused projection + bilinear sample + 3-layer dense MLP via WMMA.
// One thread per voxel; each wave32 owns 32 voxels = two 16-row WMMA tiles.
// Writes out_latent (NV,3,P), lat5 (NV,P,3) and latm (P,3) (= mean over views).
// ---------------------------------------------------------------------------
__global__ __launch_bounds__(256) void encode_kernel(
    const float* __restrict__ images, const float* __restrict__ poses,
    const float* __restrict__ focal,  const float* __restrict__ cvec,
    const float* __restrict__ dw1, const float* __restrict__ db1,
    const float* __restrict__ dw2, const float* __restrict__ db2,
    const float* __restrict__ dw3, const float* __restrict__ db3,
    float* __restrict__ out_latent, float* __restrict__ lat5,
    float* __restrict__ latm)
{
  __shared__ float xt[8][32][12];   // per-wave staging of x-vectors (K padded to 12)
  __shared__ float ht[8][32][4];    // per-wave staging of hidden activations

  const int  tid  = threadIdx.x;
  const int  wv   = tid >> 5;
  const int  lane = tid & 31;
  const int  p    = blockIdx.x * 256 + tid;     // voxel id (grid is exact multiple)
  const int  n    = lane & 15;                  // WMMA column / row-in-tile
  const bool hihl = lane >= 16;
  const int  khalf = hihl ? 2 : 0;

  // voxel world coords: p = x*96*96 + y*96 + z ; xs,ys in [-1,1], zs in [0,1]
  const int gx = p / (GSZ*GSZ);
  const int gy = (p / GSZ) % GSZ;
  const int gz = p % GSZ;
  const float wpx = -1.f + 2.f * (float)gx / (float)(GSZ - 1);
  const float wpy = -1.f + 2.f * (float)gy / (float)(GSZ - 1);
  const float wpz = (float)gz / (float)(GSZ - 1);

  // --- per-lane B-matrix fragments (K x N, N = lane&15, K striped reg x half) ---
  // layer1: W1T[k][n] = dw1[n*9+k], K padded 9->12, three K=4 chunks
  float b1x[3], b1y[3];
#pragma unroll
  for (int cc = 0; cc < 3; ++cc) {
    const int kx = 4*cc + khalf, ky = kx + 1;
    b1x[cc] = (n < 3 && kx < 9) ? dw1[n*9 + kx] : 0.f;
    b1y[cc] = (n < 3 && ky < 9) ? dw1[n*9 + ky] : 0.f;
  }
  const float b2x = (n < 3 && khalf   < 3) ? dw2[n*3 + khalf]     : 0.f;
  const float b2y = (n < 3 && khalf+1 < 3) ? dw2[n*3 + khalf + 1] : 0.f;
  const float b3x = (n < 3 && khalf   < 3) ? dw3[n*3 + khalf]     : 0.f;
  const float b3y = (n < 3 && khalf+1 < 3) ? dw3[n*3 + khalf + 1] : 0.f;
  const float cb1 = (n < 3) ? db1[n] : 0.f;   // bias along columns (all 16 rows)
  const float cb2 = (n < 3) ? db2[n] : 0.f;
  const float cb3 = (n < 3) ? db3[n] : 0.f;

  const float fpix = focal[0] * 0.5f;         // stride = 2
  const float c2x  = cvec[0] * 0.5f;
  const float c2y  = cvec[1] * 0.5f;

  // zero K-padding columns once (never overwritten)
  xt[wv][lane][9] = 0.f; xt[wv][lane][10] = 0.f; xt[wv][lane][11] = 0.f;

  float accm0 = 0.f, accm1 = 0.f, accm2 = 0.f;

  for (int v = 0; v < NVIEW; ++v) {
    __builtin_prefetch(poses + ((v + 1) & (NVIEW - 1)) * 16, 0, 1);
    const float* pv = poses + v * 16;

    // d = wg - t ; cam_i = sum_j d_j * R[j][i]
    const float d0 = wpx - pv[3];
    const float d1 = wpy - pv[7];
    const float d2 = wpz - pv[11];
    const float cam0 = d0*pv[0] + d1*pv[4] + d2*pv[8];
    const float cam1 = d0*pv[1] + d1*pv[5] + d2*pv[9];
    const float cam2 = d0*pv[2] + d1*pv[6] + d2*pv[10];

    const float e0 = d0 + 1e-9f, e1 = d1 + 1e-9f, e2 = d2 + 1e-9f;
    const float invn = 1.f / sqrtf(e0*e0 + e1*e1 + e2*e2);
    const float mz   = (cam2 < 0.001f) ? 1.f : 0.f;
    const float dir0 = d0 * invn * mz;
    const float dir1 = d1 * invn * mz;
    const float dir2 = d2 * invn * mz;

    const float invz = 1.f / cam2;
    const float u  = cam0 * invz * fpix + c2x;          // f2 = [f, -f]
    const float vq = cam1 * invz * (-fpix) + c2y;
    const float gxn = 2.f * u  / 127.f - 1.f;           // W_feat-1 = 127
    const float gyn = 2.f * vq / 127.f - 1.f;
    const float inside = (fabsf(gxn) <= 1.f && fabsf(gyn) <= 1.f && cam2 < 0.001f) ? 1.f : 0.f;

    // bilinear on full-res feats (256x256), feats = 0.5*img + 0.5
    const float ixf = (gxn + 1.f) * 127.5f;
    const float iyf = (gyn + 1.f) * 127.5f;
    const float x0f = floorf(ixf), y0f = floorf(iyf);
    const float wx  = ixf - x0f,  wy  = iyf - y0f;
    const float vx0 = (x0f >= 0.f  && x0f <= 255.f) ? 1.f : 0.f;
    const float vx1 = (x0f >= -1.f && x0f <= 254.f) ? 1.f : 0.f;
    const float vy0 = (y0f >= 0.f  && y0f <= 255.f) ? 1.f : 0.f;
    const float vy1 = (y0f >= -1.f && y0f <= 254.f) ? 1.f : 0.f;
    const int xi = (int)fminf(fmaxf(x0f, -2.f), 257.f);
    const int yi = (int)fminf(fmaxf(y0f, -2.f), 257.f);
    const int cx0 = min(max(xi,     0), 255);
    const int cx1 = min(max(xi + 1, 0), 255);
    const int cy0 = min(max(yi,     0), 255);
    const int cy1 = min(max(yi + 1, 0), 255);
    const float w00 = (1.f - wx) * (1.f - wy) * vx0 * vy0;
    const float w01 = wx * (1.f - wy) * vx1 * vy0;
    const float w10 = (1.f - wx) * wy * vx0 * vy1;
    const float w11 = wx * wy * vx1 * vy1;

    float lt[3];
#pragma unroll
    for (int ch = 0; ch < 3; ++ch) {
      const float* im = images + ((size_t)(v*3 + ch)) * 256 * 256;
      const float f00 = im[cy0*256 + cx0] * 0.5f + 0.5f;
      const float f01 = im[cy0*256 + cx1] * 0.5f + 0.5f;
      const float f10 = im[cy1*256 + cx0] * 0.5f + 0.5f;
      const float f11 = im[cy1*256 + cx1] * 0.5f + 0.5f;
      lt[ch] = inside * (w00*f00 + w01*f01 + w10*f10 + w11*f11);
      out_latent[((size_t)(v*3 + ch)) * NPTS + p] = lt[ch];   // (NV,3,P), p-coalesced
    }

    // stage x = [lat, cam, dirs] into per-wave LDS (A-matrix source)
    lds_fence();   // prior WMMA A-loads of xt must retire before overwrite
    xt[wv][lane][0] = lt[0]; xt[wv][lane][1] = lt[1]; xt[wv][lane][2] = lt[2];
    xt[wv][lane][3] = cam0;  xt[wv][lane][4] = cam1;  xt[wv][lane][5] = cam2;
    xt[wv][lane][6] = dir0;  xt[wv][lane][7] = dir1;  xt[wv][lane][8] = dir2;
    lds_fence();

    // two 16-voxel WMMA tiles per wave
#pragma unroll
    for (int t = 0; t < 2; ++t) {
      const int rb   = t*16 + n;                // A row for this lane
      const int rowb = t*16 + (hihl ? 8 : 0);   // D row base for staging

      // ---- layer 1: X(16x12) @ W1T(12x16), 3 chained K=4 WMMAs ----
      v8f acc;
#pragma unroll
      for (int j = 0; j < 8; ++j) acc[j] = cb1;
#pragma unroll
      for (int cc = 0; cc < 3; ++cc) {
        v2f a;  a.x = xt[wv][rb][4*cc + khalf];  a.y = xt[wv][rb][4*cc + khalf + 1];
        v2f b;  b.x = b1x[cc];                   b.y = b1y[cc];
        acc = __builtin_amdgcn_wmma_f32_16x16x4_f32(false, a, false, b,
                                                    (short)0, acc, false, false);
      }
      if (n < 4) {
#pragma unroll
        for (int j = 0; j < 8; ++j) ht[wv][rowb + j][n] = fmaxf(acc[j], 0.f);
      }
      lds_fence();

      // ---- layer 2 ----
      v2f a2; a2.x = ht[wv][rb][khalf]; a2.y = ht[wv][rb][khalf + 1];
      v2f b2; b2.x = b2x;               b2.y = b2y;
      v8f acc2;
#pragma unroll
      for (int j = 0; j < 8; ++j) acc2[j] = cb2;
      acc2 = __builtin_amdgcn_wmma_f32_16x16x4_f32(false, a2, false, b2,
                                                   (short)0, acc2, false, false);
      if (n < 4) {
#pragma unroll
        for (int j = 0; j < 8; ++j) ht[wv][rowb + j][n] = fmaxf(acc2[j], 0.f);
      }
      lds_fence();

      // ---- layer 3 (no relu) ----
      v2f a3; a3.x = ht[wv][rb][khalf]; a3.y = ht[wv][rb][khalf + 1];
      v2f b3; b3.x = b3x;               b3.y = b3y;
      v8f acc3;
#pragma unroll
      for (int j = 0; j < 8; ++j) acc3[j] = cb3;
      acc3 = __builtin_amdgcn_wmma_f32_16x16x4_f32(false, a3, false, b3,
                                                   (short)0, acc3, false, false);
      if (n < 4) {
#pragma unroll
        for (int j = 0; j < 8; ++j) ht[wv][rowb + j][n] = acc3[j];
      }
      lds_fence();
    }

    // each lane reads its own voxel's h3 and writes lat5 (coalesced 3-float runs)
    const float h0 = ht[wv][lane][0];
    const float h1 = ht[wv][lane][1];
    const float h2 = ht[wv][lane][2];
    float* l5 = lat5 + ((size_t)v * NPTS + p) * 3;
    l5[0] = h0; l5[1] = h1; l5[2] = h2;
    accm0 += h0; accm1 += h1; accm2 += h2;
  }

  latm[(size_t)p*3 + 0] = accm0 * 0.125f;
  latm[(size_t)p*3 + 1] = accm1 * 0.125f;
  latm[(size_t)p*3 + 2] = accm2 * 0.125f;
}

// ---------------------------------------------------------------------------
// Kernel B: per-plane axis-softmax aggregation.
// plane 0: reduce over X -> scene_yz ; 1: over Y -> scene_xz ; 2: over Z -> scene_xy
// One thread per output pixel; two-pass numerically-stable softmax over 96.
// ---------------------------------------------------------------------------
__global__ __launch_bounds__(256) void agg_kernel(
    const float* __restrict__ latm,
    const float* __restrict__ w1, const float* __restrict__ b1,
    const float* __restrict__ w2, const float* __restrict__ b2,
    float* __restrict__ outp, int plane)
{
  const int idx = blockIdx.x * 256 + threadIdx.x;
  if (idx >= PLANE) return;
  const int i = idx / GSZ, j = idx % GSZ;

  float W1[12], B1[3], W2[3];
#pragma unroll
  for (int k = 0; k < 12; ++k) W1[k] = w1[k];
#pragma unroll
  for (int k = 0; k < 3; ++k) { B1[k] = b1[k]; W2[k] = w2[k]; }
  const float B2 = b2[0];

  size_t base, stride;
  float c0, cstep;
  if (plane == 0) {        // fix (y=i, z=j), loop x
    base = ((size_t)i * GSZ + j) * 3;  stride = (size_t)GSZ * GSZ * 3;
    c0 = -1.f; cstep = 2.f / (GSZ - 1);
  } else if (plane == 1) { // fix (x=i, z=j), loop y
    base = ((size_t)i * GSZ * GSZ + j) * 3;  stride = (size_t)GSZ * 3;
    c0 = -1.f; cstep = 2.f / (GSZ - 1);
  } else {                 // fix (x=i, y=j), loop z
    base = ((size_t)i * GSZ + j) * (size_t)GSZ * 3;  stride = 3;
    c0 = 0.f;  cstep = 1.f / (GSZ - 1);
  }

  auto score = [&](float l0, float l1, float l2, float co) -> float {
    const float h0 = fmaxf(W1[0]*l0 + W1[1]*l1 + W1[2]*l2  + W1[3]*co  + B1[0], 0.f);
    const float h1 = fmaxf(W1[4]*l0 + W1[5]*l1 + W1[6]*l2  + W1[7]*co  + B1[1], 0.f);
    const float h2 = fmaxf(W1[8]*l0 + W1[9]*l1 + W1[10]*l2 + W1[11]*co + B1[2], 0.f);
    return W2[0]*h0 + W2[1]*h1 + W2[2]*h2 + B2;
  };

  float smax = -1e30f;
  for (int r = 0; r < GSZ; ++r) {
    const float* lp = latm + base + (size_t)r * stride;
    smax = fmaxf(smax, score(lp[0], lp[1], lp[2], c0 + cstep * r));
  }
  float den = 0.f, n0 = 0.f, n1 = 0.f, n2 = 0.f;
  for (int r = 0; r < GSZ; ++r) {
    const float* lp = latm + base + (size_t)r * stride;
    const float l0 = lp[0], l1 = lp[1], l2 = lp[2];
    const float e = expf(score(l0, l1, l2, c0 + cstep * r) - smax);
    den += e; n0 += e*l0; n1 += e*l1; n2 += e*l2;
  }
  const float inv = 1.f / den;
  outp[0*PLANE + idx] = n0 * inv;
  outp[1*PLANE + idx] = n1 * inv;
  outp[2*PLANE + idx] = n2 * inv;
}

// ---------------------------------------------------------------------------
extern "C" void kernel_launch(void* const* d_in, const int* in_sizes, int n_in,
                              void* d_out, int out_size, void* d_ws, size_t ws_size,
                              hipStream_t stream) {
  const float* images = (const float*)d_in[0];
  const float* poses  = (const float*)d_in[1];
  const float* focal  = (const float*)d_in[2];
  const float* cvec   = (const float*)d_in[3];
  const float* dw1 = (const float*)d_in[4];
  const float* db1 = (const float*)d_in[5];
  const float* dw2 = (const float*)d_in[6];
  const float* db2 = (const float*)d_in[7];
  const float* dw3 = (const float*)d_in[8];
  const float* db3 = (const float*)d_in[9];
  const float* ayz_w1 = (const float*)d_in[10];
  const float* ayz_b1 = (const float*)d_in[11];
  const float* ayz_w2 = (const float*)d_in[12];
  const float* ayz_b2 = (const float*)d_in[13];
  const float* axz_w1 = (const float*)d_in[14];
  const float* axz_b1 = (const float*)d_in[15];
  const float* axz_w2 = (const float*)d_in[16];
  const float* axz_b2 = (const float*)d_in[17];
  const float* axy_w1 = (const float*)d_in[18];
  const float* axy_b1 = (const float*)d_in[19];
  const float* axy_w2 = (const float*)d_in[20];
  const float* axy_b2 = (const float*)d_in[21];

  float* out = (float*)d_out;
  // return order: scene_xz, scene_xy, scene_yz, out_latent, lat5
  float* scene_xz   = out;
  float* scene_xy   = out + 3 * PLANE;
  float* scene_yz   = out + 6 * PLANE;
  float* out_latent = out + 9 * PLANE;
  float* lat5       = out_latent + (size_t)NVIEW * NPTS * 3;
  float* latm       = (float*)d_ws;            // P*3 floats (~10.6 MB)

  encode_kernel<<<NPTS / 256, 256, 0, stream>>>(
      images, poses, focal, cvec, dw1, db1, dw2, db2, dw3, db3,
      out_latent, lat5, latm);

  const int ab = (PLANE + 255) / 256;
  agg_kernel<<<ab, 256, 0, stream>>>(latm, ayz_w1, ayz_b1, ayz_w2, ayz_b2, scene_yz, 0);
  agg_kernel<<<ab, 256, 0, stream>>>(latm, axz_w1, axz_b1, axz_w2, axz_b2, scene_xz, 1);
  agg_kernel<<<ab, 256, 0, stream>>>(latm, axy_w1, axy_b1, axy_w2, axy_b2, scene_xy, 2);
}